// ExemplarModel_15882789060827
// MI455X (gfx1250) — compile-verified
//
#include <hip/hip_runtime.h>
#include <hip/hip_bf16.h>

// Problem constants (match reference)
#define B_  32
#define N_  32768
#define D_  128
#define C_  64

#define CHUNKS_PER_B 64                 // workgroups per batch row
#define BLOCK 32                        // single wave32 per workgroup
#define ROWS 16                         // one WMMA M-tile per iteration
#define N_PER_WG (N_ / CHUNKS_PER_B)    // 512
#define ITERS (N_PER_WG / ROWS)         // 32
#define TSTRIDE 132                     // padded LDS row stride (floats)

typedef float v2f __attribute__((ext_vector_type(2)));
typedef float v8f __attribute__((ext_vector_type(8)));
typedef int   v4i __attribute__((vector_size(4 * sizeof(int))));   // matches builtin param
typedef __attribute__((address_space(1))) v4i* gv4i_p;
typedef __attribute__((address_space(3))) v4i* lv4i_p;

// Stage one 16x128 fp32 tile (row-major, row stride D_) into LDS with padded
// stride. Preferred path: CDNA5 async global->LDS (ASYNCcnt-tracked), which
// keeps the copy out of VGPRs and overlaps with WMMA on the other buffer.
__device__ __forceinline__ void stage_tile(const float* __restrict__ gsrc,
                                           float* ldst, int lane)
{
#if __has_builtin(__builtin_amdgcn_global_load_async_to_lds_b128)
#pragma unroll
    for (int r = 0; r < ROWS; ++r) {
        __builtin_amdgcn_global_load_async_to_lds_b128(
            (gv4i_p)(gsrc + (size_t)r * D_ + lane * 4),
            (lv4i_p)(ldst + r * TSTRIDE + lane * 4),
            /*offset=*/0, /*cpol=*/0);
    }
#else
#pragma unroll
    for (int r = 0; r < ROWS; ++r) {
        float4 v = ((const float4*)(gsrc + (size_t)r * D_))[lane];
        float* d = ldst + r * TSTRIDE + lane * 4;
        d[0] = v.x; d[1] = v.y; d[2] = v.z; d[3] = v.w;
    }
#endif
}

__global__ __launch_bounds__(BLOCK) void exemplar_pass1(
    const float* __restrict__ probes,   // [B,D]
    const float* __restrict__ emb,      // [B,N,D]
    const int*   __restrict__ seg,      // [B,N] sorted per row
    const float* __restrict__ kw,       // [1]
    float*       __restrict__ partial)  // [B][CHUNKS_PER_B][C]
{
    __shared__ float p_s[D_];
    __shared__ float tile[2][ROWS * TSTRIDE];   // double buffer
    __shared__ float ee_s[ROWS];
    __shared__ float acts_s[ROWS];
    __shared__ float bins[C_];

    const int lane  = threadIdx.x;
    const int b     = blockIdx.x / CHUNKS_PER_B;
    const int chunk = blockIdx.x % CHUNKS_PER_B;
    const int hi    = lane >> 4;        // half-wave select
    const int row16 = lane & 15;
    const int koff  = hi * 2;           // A/B fragment K sub-offset

    // Stage probe row (32 lanes x float4 = 128 floats); zero class bins.
    {
        float4 v = ((const float4*)(probes + (size_t)b * D_))[lane];
        float* d = &p_s[lane * 4];
        d[0] = v.x; d[1] = v.y; d[2] = v.z; d[3] = v.w;
    }
    bins[lane]      = 0.0f;
    bins[lane + 32] = 0.0f;
    __syncthreads();                    // single wave: compiler/mem ordering

    // ||p||^2 (redundant per lane, trivial)
    float pp = 0.0f;
    const float4* p4 = (const float4*)p_s;
#pragma unroll
    for (int k = 0; k < D_ / 4; ++k) {
        float4 v = p4[k];
        pp += v.x * v.x + v.y * v.y + v.z * v.z + v.w * v.w;
    }
    const float inv_kw = 1.0f / kw[0];

    const float* gbase = emb + ((size_t)b * N_ + (size_t)chunk * N_PER_WG) * D_;

    // Prologue: fill buffer 0.
    stage_tile(gbase, tile[0], lane);

    for (int it = 0; it < ITERS; ++it) {
        float* tbuf = tile[it & 1];

        // Prefetch next tile into the other buffer, then wait for the current
        // tile: async loads complete in order, so <=16 outstanding means the
        // older 16 (current tile) have landed in LDS.
        if (it + 1 < ITERS) {
            stage_tile(gbase + (size_t)(it + 1) * ROWS * D_, tile[(it + 1) & 1], lane);
            asm volatile("s_wait_asynccnt 0x10" ::: "memory");
        } else {
            asm volatile("s_wait_asynccnt 0x0" ::: "memory");
        }

        // --- ||e||^2 per row: 2 lanes/row, 64 cols each, shfl combine ---
        {
            const int r  = lane >> 1;
            const int cb = (lane & 1) * 64;
            float ee = 0.0f;
#pragma unroll
            for (int k = 0; k < 16; ++k) {
                const float* q = &tbuf[r * TSTRIDE + cb + 4 * k];
                ee += q[0] * q[0] + q[1] * q[1] + q[2] * q[2] + q[3] * q[3];
            }
            ee += __shfl_xor(ee, 1, 32);
            if ((lane & 1) == 0) ee_s[r] = ee;
        }
        asm volatile("" ::: "memory");

        // --- p·e for 16 exemplars: chained V_WMMA_F32_16X16X4_F32 over K=128.
        // A (16x4): lanes0-15 row=lane K=4j..4j+1; lanes16-31 K=4j+2..4j+3.
        // B (4x16): probe values broadcast across all 16 columns.
        v8f c = {0.f, 0.f, 0.f, 0.f, 0.f, 0.f, 0.f, 0.f};
#pragma unroll
        for (int j = 0; j < 32; ++j) {
            v2f a  = *(const v2f*)&tbuf[row16 * TSTRIDE + 4 * j + koff];
            v2f bb = *(const v2f*)&p_s[4 * j + koff];
            c = __builtin_amdgcn_wmma_f32_16x16x4_f32(
                    false, a, false, bb, (short)0, c, false, false);
        }

        // --- acts: C layout => lanes0-15 rows 0-7 in c[0..7], lanes16-31 rows 8-15.
#pragma unroll
        for (int i = 0; i < 8; ++i) {
            const int e  = i + 8 * hi;
            float pe  = c[i];
            float ee  = ee_s[e];
            float d2  = fmaxf(pp + ee - 2.0f * pe, 0.0f);
            float act = __expf(-sqrtf(d2) * inv_kw);
            if (row16 == 0) acts_s[e] = act;   // lanes 0 and 16 publish
        }
        asm volatile("" ::: "memory");

        // --- deterministic binning: lane 0, fixed order ---
        if (lane == 0) {
            const int n_base = chunk * N_PER_WG + it * ROWS;
#pragma unroll
            for (int e = 0; e < ROWS; ++e) {
                int s = seg[(size_t)b * N_ + n_base + e];
                bins[s] += acts_s[e];
            }
        }
        asm volatile("" ::: "memory");
    }

    // Per-(b,chunk) partial sums (wave-private bins, no atomics).
    float* dst = partial + ((size_t)b * CHUNKS_PER_B + chunk) * C_;
    dst[lane]      = bins[lane];
    dst[lane + 32] = bins[lane + 32];
}

__global__ void exemplar_pass2(const int* __restrict__ seg,
                               const float* __restrict__ partial,
                               float* __restrict__ out)
{
    int id = blockIdx.x * blockDim.x + threadIdx.x;
    if (id >= B_ * C_) return;
    int b = id / C_;
    int c = id % C_;

    float s = 0.0f;
    for (int k = 0; k < CHUNKS_PER_B; ++k)
        s += partial[((size_t)b * CHUNKS_PER_B + k) * C_ + c];

    // Counts via binary search on the sorted per-row segment ids.
    const int* ids = seg + (size_t)b * N_;
    int lo = 0, hiN = N_;
    while (lo < hiN) { int m = (lo + hiN) >> 1; if (ids[m] < c) lo = m + 1; else hiN = m; }
    int lb = lo;
    hiN = N_;
    while (lo < hiN) { int m = (lo + hiN) >> 1; if (ids[m] < c + 1) lo = m + 1; else hiN = m; }
    int cnt = lo - lb;

    out[id] = s / fmaxf((float)cnt, 1.0f);
}

extern "C" void kernel_launch(void* const* d_in, const int* in_sizes, int n_in,
                              void* d_out, int out_size, void* d_ws, size_t ws_size,
                              hipStream_t stream) {
    const float* probes = (const float*)d_in[0];
    const float* emb    = (const float*)d_in[1];
    const int*   seg    = (const int*)d_in[2];
    const float* kw     = (const float*)d_in[3];
    float* out     = (float*)d_out;
    float* partial = (float*)d_ws;   // B*CHUNKS_PER_B*C floats = 512 KB

    dim3 g1(B_ * CHUNKS_PER_B);      // 2048 single-wave workgroups
    exemplar_pass1<<<g1, BLOCK, 0, stream>>>(probes, emb, seg, kw, partial);

    dim3 g2((B_ * C_ + 255) / 256);
    exemplar_pass2<<<g2, 256, 0, stream>>>(seg, partial, out);
}